// SparseCodingLoss_42606075576971
// MI455X (gfx1250) — compile-verified
//
#include <hip/hip_runtime.h>
#include <hip/hip_bf16.h>

// Matching-pursuit sparse coding loss for MI455X (gfx1250, wave32).
// Full fp32 WMMA conv once, then per-step incremental window re-conv
// (only t in [ti-127, ti+127] changes) + 2-phase u64-key argmax.
// fm (~33MB) stays L2-resident on the 192MB L2.
// fm is padded by ROW_PAD t-positions on both ends of every row so the
// WMMA C-tile stores need no bounds check (no exec-mask branches).
// Dictionary stored K-pair-packed (dP) so each WMMA B operand is one
// coalesced global_load_b64.
// Requires ws_size >= ~34 MB.

#define T_LEN 16384
#define A_N   128
#define K_LEN 128
#define N_STEPS 16
#define E_DIM 126
#define NROW  4          // rows: a-b0, a-b1, b-b0, b-b1
#define CHUNK 8192
#define NCHUNK ((T_LEN * A_N) / CHUNK)   // 256
#define ROW_PAD 160      // padded t-positions each side (window reach <= 144)
#define FM_STRIDE ((size_t)(T_LEN + 2 * ROW_PAD) * A_N)   // floats per row

typedef __attribute__((ext_vector_type(2))) float v2f;
typedef __attribute__((ext_vector_type(8))) float v8f;

// ---------------------------------------------------------------- d_unit
// dP layout: for k-pair p = k/2, dP[p*A_N + a] = {dunit[a][2p], dunit[a][2p+1]}
__global__ __launch_bounds__(128) void k_dunit(const float* __restrict__ d,
                                               float* __restrict__ dunit,
                                               float2* __restrict__ dP) {
  int a = blockIdx.x;            // atom
  int k = threadIdx.x;           // tap
  __shared__ float sh[128];
  float v = d[a * K_LEN + k];
  sh[k] = v * v;
  __syncthreads();
  for (int off = 64; off; off >>= 1) {
    if (k < off) sh[k] += sh[k + off];
    __syncthreads();
  }
  float u = v * (1.0f / (sqrtf(sh[0]) + 1e-8f));
  dunit[a * K_LEN + k] = u;      // row-major (for residual update)
  ((float*)dP)[((k >> 1) * A_N + a) * 2 + (k & 1)] = u;   // K-pair packed
}

// ---------------------------------------------------------------- res init
__global__ __launch_bounds__(256) void k_init_res(const float* __restrict__ a,
                                                  const float* __restrict__ b,
                                                  float* __restrict__ res) {
  int row = blockIdx.y;
  int i = blockIdx.x * 256 + threadIdx.x;
  const float* src = (row < 2) ? a : b;
  res[row * T_LEN + i] = src[(row & 1) * T_LEN + i];
}

// ------------------------------------------------- WMMA conv tile (16t x 16a)
// fm[t][a] = sum_k res[t + k - 64] * dunit[a][k], fp32 WMMA 16x16x4.
// f points at the valid-t0 origin of the row; stores may touch the pad.
__device__ __forceinline__ void conv_tile_store(const float* __restrict__ sres,
                                                const float2* __restrict__ dP,
                                                float* __restrict__ f,
                                                int t0, int a0) {
  const int lane = threadIdx.x & 31;
  const int half = lane >> 4;     // K-pair select (A: VGPR0=K(2h), VGPR1=K(2h+1))
  const int l16  = lane & 15;     // A: M row / B: N col
  const float2* dp = dP + (a0 + l16);   // + pair*A_N per step
  v8f acc = {};
#pragma unroll 8
  for (int kc = 0; kc < K_LEN; kc += 4) {
    const int k0 = kc + half * 2;
    v2f Av, Bv;
    // A (16x4 fp32): sres[x] = res[t0 - 64 + x], x = l16 + k0 in [0,142]
    Av.x = sres[l16 + k0];
    Av.y = sres[l16 + k0 + 1];
    // B (4x16 fp32): one b64 load: {dunit[a][k0], dunit[a][k0+1]}
    float2 bp = dp[(k0 >> 1) * A_N];
    Bv.x = bp.x;
    Bv.y = bp.y;
    acc = __builtin_amdgcn_wmma_f32_16x16x4_f32(
        false, Av, false, Bv, (short)0, acc, false, false);
  }
  // C/D: VGPR r -> M=r (lanes 0-15) / M=r+8 (lanes 16-31); N = l16
  // Unguarded: the fm row is padded by ROW_PAD t-positions on each side.
  float* fb = f + ((long long)(t0 + 8 * half) * A_N + (a0 + l16));
#pragma unroll
  for (int v = 0; v < 8; ++v) fb[(long long)v * A_N] = acc[v];
}

// mode 0: full conv, grid (T/16, NROW). mode 1: window re-conv around last
// pick, grid (16, NROW), t0 = tstart[row] + 16*bx (covers ti-127..ti+128).
__global__ __launch_bounds__(256) void k_conv(const float* __restrict__ res,
                                              const float2* __restrict__ dP,
                                              float* __restrict__ fm,
                                              const int* __restrict__ tstart,
                                              int mode) {
  __shared__ float sres[144];
  const int row = blockIdx.y;
  const int t0 = mode ? (tstart[row] + (int)blockIdx.x * 16)
                      : ((int)blockIdx.x * 16);
  const float* r = res + row * T_LEN;
  const int tid = threadIdx.x;
  if (tid < 144) {                       // stage residual window (zero-padded)
    int g = t0 - (K_LEN / 2) + tid;
    int ok = (g >= 0) & (g < T_LEN);
    sres[tid] = ok ? r[g] : 0.0f;
  }
  __syncthreads();                       // EXEC all-ones past here for WMMA
  conv_tile_store(sres, dP, fm + (size_t)row * FM_STRIDE, t0,
                  (tid >> 5) * 16);
}

// ---------------------------------------------------------------- argmax p1
__global__ __launch_bounds__(256) void k_argmax_part(
    const float* __restrict__ fm, unsigned long long* __restrict__ part) {
  const int row = blockIdx.y;
  const int base = blockIdx.x * CHUNK;
  const float4* f4 =
      reinterpret_cast<const float4*>(fm + (size_t)row * FM_STRIDE + base);
  unsigned long long best = 0;
#pragma unroll 4
  for (int i = threadIdx.x; i < CHUNK / 4; i += 256) {
    float4 v = f4[i];
    const int idx = base + i * 4;
    float vv[4] = {v.x, v.y, v.z, v.w};
#pragma unroll
    for (int j = 0; j < 4; ++j) {
      unsigned u = __float_as_uint(vv[j]);
      u = (u & 0x80000000u) ? ~u : (u | 0x80000000u);   // monotone key
      unsigned long long key =
          ((unsigned long long)u << 32) | (unsigned)(idx + j);
      best = (key > best) ? key : best;
    }
  }
  __shared__ unsigned long long s[256];
  s[threadIdx.x] = best;
  __syncthreads();
  for (int off = 128; off; off >>= 1) {
    if (threadIdx.x < off) {
      unsigned long long o = s[threadIdx.x + off];
      if (o > s[threadIdx.x]) s[threadIdx.x] = o;
    }
    __syncthreads();
  }
  if (threadIdx.x == 0) part[row * NCHUNK + blockIdx.x] = s[0];
}

// ------------------------------------------- argmax p2 + emb + res update
__global__ __launch_bounds__(256) void k_finalize(
    const unsigned long long* __restrict__ part,
    const float* __restrict__ dunit, const float* __restrict__ atom_emb,
    float* __restrict__ res, float* __restrict__ embs,
    int* __restrict__ tstart, int step) {
  const int row = blockIdx.x;
  __shared__ unsigned long long s[256];
  s[threadIdx.x] = part[row * NCHUNK + threadIdx.x];
  __syncthreads();
  for (int off = 128; off; off >>= 1) {
    if (threadIdx.x < off) {
      unsigned long long o = s[threadIdx.x + off];
      if (o > s[threadIdx.x]) s[threadIdx.x] = o;
    }
    __syncthreads();
  }
  const unsigned long long key = s[0];
  const int idx = (int)(key & 0xFFFFFFFFull);
  unsigned ub = (unsigned)(key >> 32);
  unsigned fb = (ub & 0x80000000u) ? (ub ^ 0x80000000u) : ~ub;
  const float value = __uint_as_float(fb);
  const int a = idx & (A_N - 1);       // fm layout [t][a]
  const int t = idx / A_N;
  const int a_eff = (value > 0.0f) ? a : 0;  // soft_dirac forward == one-hot
  const int t_eff = (value > 0.0f) ? t : 0;  // of argmax of sparse row
  float* e = embs + ((size_t)row * N_STEPS + step) * 128;
  if (threadIdx.x == 0) {
    e[0] = ((float)t_eff / (float)(T_LEN - 1)) * 20.0f;  // rng[ti]*20
    e[1] = value;
    tstart[row] = t - 127;             // window base for re-conv
  }
  if (threadIdx.x < E_DIM)
    e[2 + threadIdx.x] = atom_emb[a_eff * E_DIM + threadIdx.x];
  if (threadIdx.x < K_LEN) {           // res -= value * d_unit[a] at t-64..t+63
    int ri = t + threadIdx.x - (K_LEN / 2);
    if (ri >= 0 && ri < T_LEN)
      res[row * T_LEN + ri] -= value * dunit[a * K_LEN + threadIdx.x];
  }
}

// ---------------------------------------------------------------- norms
__global__ __launch_bounds__(256) void k_norms(const float* __restrict__ res,
                                               float* __restrict__ norms) {
  const int row = blockIdx.x;
  float acc = 0.f;
  for (int i = threadIdx.x; i < T_LEN; i += 256) {
    float v = res[row * T_LEN + i];
    acc += v * v;
  }
  __shared__ float sh[256];
  sh[threadIdx.x] = acc;
  __syncthreads();
  for (int off = 128; off; off >>= 1) {
    if (threadIdx.x < off) sh[threadIdx.x] += sh[threadIdx.x + off];
    __syncthreads();
  }
  if (threadIdx.x == 0) norms[row] = sqrtf(sh[0]);
}

// --------------------------------------------- canonical order + final loss
__global__ __launch_bounds__(256) void k_loss(const float* __restrict__ embs,
                                              const float* __restrict__ proj,
                                              const float* __restrict__ norms,
                                              float* __restrict__ out) {
  __shared__ float keys[NROW][N_STEPS];
  __shared__ int order[NROW][N_STEPS];
  __shared__ float red[256];
  const int tid = threadIdx.x;
  if (tid < NROW * N_STEPS) {               // keys = embs @ proj
    int row = tid / N_STEPS, s = tid % N_STEPS;
    const float* e = embs + ((size_t)row * N_STEPS + s) * 128;
    float k = 0.f;
    for (int j = 0; j < 128; ++j) k += e[j] * proj[j];
    keys[row][s] = k;
  }
  __syncthreads();
  if (tid < NROW) {                         // stable ascending argsort (16)
    unsigned used = 0;
    for (int r = 0; r < N_STEPS; ++r) {
      int bi = -1;
      float bv = 0.f;
      for (int s = 0; s < N_STEPS; ++s) {
        if (used & (1u << s)) continue;
        if (bi < 0 || keys[tid][s] < bv) { bi = s; bv = keys[tid][s]; }
      }
      order[tid][r] = bi;
      used |= 1u << bi;
    }
  }
  __syncthreads();
  float acc = 0.f;                          // mean((ae-be)^2)
  for (int i = tid; i < 2 * N_STEPS * 128; i += 256) {
    int batch = i / (N_STEPS * 128);
    int s = (i / 128) % N_STEPS;
    int j = i % 128;
    int ra = batch, rb = 2 + batch;
    float av = embs[((size_t)ra * N_STEPS + order[ra][s]) * 128 + j];
    float bv = embs[((size_t)rb * N_STEPS + order[rb][s]) * 128 + j];
    float d = av - bv;
    acc += d * d;
  }
  red[tid] = acc;
  __syncthreads();
  for (int off = 128; off; off >>= 1) {
    if (tid < off) red[tid] += red[tid + off];
    __syncthreads();
  }
  if (tid == 0) {
    float mse = red[0] / (float)(2 * N_STEPS * 128);
    float nl = 0.5f * (fabsf(norms[0] - norms[2]) + fabsf(norms[1] - norms[3]));
    out[0] = mse + nl;
  }
}

// ================================================================= launch
extern "C" void kernel_launch(void* const* d_in, const int* in_sizes, int n_in,
                              void* d_out, int out_size, void* d_ws,
                              size_t ws_size, hipStream_t stream) {
  const float* a        = (const float*)d_in[0];  // (2,1,16384)
  const float* b        = (const float*)d_in[1];  // (2,1,16384)
  const float* d        = (const float*)d_in[2];  // (128,128)
  const float* atom_emb = (const float*)d_in[3];  // (128,126)
  const float* proj     = (const float*)d_in[4];  // (128,)

  char* ws = (char*)d_ws;
  float*  dunit = (float*)(ws + 0);                //  64 KB
  float2* dP    = (float2*)(ws + 65536);           //  64 KB (K-pair packed)
  float*  res   = (float*)(ws + 131072);           // 256 KB (4 rows)
  float*  embs  = (float*)(ws + 393216);           //  32 KB (4x16x128)
  float*  norms = (float*)(ws + 425984);           //  16 B
  int*    tst   = (int*)  (ws + 426000);           //  16 B
  unsigned long long* part = (unsigned long long*)(ws + 426016);  // 8 KB
  // fm rows padded by ROW_PAD t-positions on each side; fm points at the
  // valid t=0 origin of row 0 (rows separated by FM_STRIDE floats).
  float* fm = (float*)(ws + (1 << 19)) + (size_t)ROW_PAD * A_N;   // ~33 MB

  k_dunit<<<dim3(A_N), dim3(128), 0, stream>>>(d, dunit, dP);
  k_init_res<<<dim3(T_LEN / 256, NROW), dim3(256), 0, stream>>>(a, b, res);
  // full conv: fm[row][t][a]
  k_conv<<<dim3(T_LEN / 16, NROW), dim3(256), 0, stream>>>(res, dP, fm, tst, 0);

  for (int step = 0; step < N_STEPS; ++step) {
    k_argmax_part<<<dim3(NCHUNK, NROW), dim3(256), 0, stream>>>(fm, part);
    k_finalize<<<dim3(NROW), dim3(256), 0, stream>>>(part, dunit, atom_emb,
                                                     res, embs, tst, step);
    // only t in [ti-127, ti+128] changed: 16 tiles x 128 atoms via WMMA
    k_conv<<<dim3(16, NROW), dim3(256), 0, stream>>>(res, dP, fm, tst, 1);
  }

  k_norms<<<dim3(NROW), dim3(256), 0, stream>>>(res, norms);
  k_loss<<<dim3(1), dim3(256), 0, stream>>>(embs, proj, norms, (float*)d_out);
}